// NTM_11948599017545
// MI455X (gfx1250) — compile-verified
//
#include <hip/hip_runtime.h>
#include <math.h>

typedef __attribute__((ext_vector_type(2))) float v2f;
typedef __attribute__((ext_vector_type(8))) float v8f;

#define WCOLS 272            // fused output columns (16 * 17 tiles), all fields 16B-aligned
#define NTILES 17
#define ROWS_PER_BLOCK 128   // 4 waves x 2 groups x 16 rows
#define VSTRIDE 276          // LDS row stride; 276 % 64 = 20 -> b128 lane windows tile banks 2x (optimal)

// Fused column map (all vector fields 16B aligned):
//  [0,6) c | 6 beta_r | 7 g_r | [8,28) k_r | [28,48) k_w | 48 beta_w | 49 g_w |
//  [50,52) pad | [52,252) e_pre | [252,272) a
#define COL_BETA_R 6
#define COL_G_R    7
#define COL_KR     8
#define COL_KW     28
#define COL_BETA_W 48
#define COL_G_W    49
#define COL_E      52
#define COL_A      252

__device__ __forceinline__ float sigm(float v)      { return 1.0f / (1.0f + __expf(-v)); }
__device__ __forceinline__ float softplusf(float v) { return (v > 20.0f) ? v : log1pf(__expf(v)); }
__device__ __forceinline__ float4 lds4(const float* p) { return *(const float4*)p; }

// ---------------------------------------------------------------------------
// Setup kernel: fold  c = x@Wc+bc  into every downstream linear layer and emit
// the 8x272 fused panel PRE-SWIZZLED into WMMA B-operand layout:
//   slot (tile j, lane lo)    = {W[0][col], W[1][col], W[4][col], W[5][col]}
//   slot (tile j, lane 16+lo) = {W[2][col], W[3][col], bias[col], 0}
// (row 6 = bias is consumed via homogeneous x[6]=1, row 7 = 0)
// One ds_load_b128 per lane per tile fetches both K-step B fragments.
// ---------------------------------------------------------------------------
__global__ void ntm_fold_weights(
    const float* __restrict__ Wc,  const float* __restrict__ bc,
    const float* __restrict__ rWk, const float* __restrict__ rbk,
    const float* __restrict__ rWb, const float* __restrict__ rbb,
    const float* __restrict__ rWg, const float* __restrict__ rbg,
    const float* __restrict__ wWk, const float* __restrict__ wbk,
    const float* __restrict__ wWb, const float* __restrict__ wbb,
    const float* __restrict__ wWg, const float* __restrict__ wbg,
    const float* __restrict__ We,  const float* __restrict__ be,
    const float* __restrict__ Wa,  const float* __restrict__ ba,
    float4* __restrict__ Wpacked)      // NTILES * 32 float4 slots
{
    int col = blockIdx.x * blockDim.x + threadIdx.x;
    if (col >= WCOLS) return;

    float w[6];
    float bias;
    if (col < 6) {
        #pragma unroll
        for (int i = 0; i < 6; ++i) w[i] = Wc[i * 6 + col];
        bias = bc[col];
    } else if (col == 50 || col == 51) {
        #pragma unroll
        for (int i = 0; i < 6; ++i) w[i] = 0.0f;
        bias = 0.0f;
    } else {
        const float* Wp; const float* bp; int n; int m;
        if      (col == COL_BETA_R) { Wp = rWb; bp = rbb; n = 1;   m = 0; }
        else if (col == COL_G_R)    { Wp = rWg; bp = rbg; n = 1;   m = 0; }
        else if (col < COL_KW)      { Wp = rWk; bp = rbk; n = 20;  m = col - COL_KR; }
        else if (col < COL_BETA_W)  { Wp = wWk; bp = wbk; n = 20;  m = col - COL_KW; }
        else if (col == COL_BETA_W) { Wp = wWb; bp = wbb; n = 1;   m = 0; }
        else if (col == COL_G_W)    { Wp = wWg; bp = wbg; n = 1;   m = 0; }
        else if (col < COL_A)       { Wp = We;  bp = be;  n = 200; m = col - COL_E; }
        else                        { Wp = Wa;  bp = ba;  n = 20;  m = col - COL_A; }
        float colw[6];
        #pragma unroll
        for (int j = 0; j < 6; ++j) colw[j] = Wp[j * n + m];
        #pragma unroll
        for (int i = 0; i < 6; ++i) {
            float s = 0.0f;
            #pragma unroll
            for (int j = 0; j < 6; ++j) s += Wc[i * 6 + j] * colw[j];
            w[i] = s;
        }
        float bs = bp[m];
        #pragma unroll
        for (int j = 0; j < 6; ++j) bs += bc[j] * colw[j];
        bias = bs;
    }

    const int j  = col >> 4;
    const int lo = col & 15;
    float4 q0; q0.x = w[0]; q0.y = w[1]; q0.z = w[4]; q0.w = w[5];
    float4 q1; q1.x = w[2]; q1.y = w[3]; q1.z = bias; q1.w = 0.0f;
    Wpacked[j * 32 + lo]      = q0;   // lanes 0-15: K rows {0,1} and {4,5}
    Wpacked[j * 32 + 16 + lo] = q1;   // lanes 16-31: K rows {2,3} and {6,7}
}

// ---------------------------------------------------------------------------
// Main kernel: fused GEMM (f32 WMMA 16x16x4) -> LDS transpose -> per-row
// epilogue (cosine addressing, softmax, gating, memory update, output fc).
// ---------------------------------------------------------------------------
__global__ void __launch_bounds__(ROWS_PER_BLOCK) ntm_main(
    const float* __restrict__ x,
    const float* __restrict__ prev_read,
    const float* __restrict__ mem,
    const float4* __restrict__ Wpacked,
    const float* __restrict__ Wfc,
    const float* __restrict__ bfc,
    float* __restrict__ out, int Btot)
{
    extern __shared__ float smem[];
    float4* ldsB = (float4*)smem;                 // NTILES*32 B-operand slots (8.7 KB)
    float*  vals = smem + 4 * NTILES * 32;        // 128 rows x VSTRIDE activations

    const int t       = threadIdx.x;
    const int rowBase = blockIdx.x * ROWS_PER_BLOCK;

    // Stage pre-swizzled B panel into LDS.
    for (int i = t; i < NTILES * 32; i += ROWS_PER_BLOCK) ldsB[i] = Wpacked[i];

    // Warm this thread's memory row for the epilogue (global_prefetch_b8).
    {
        int r = min(rowBase + t, Btot - 1);
        const float* mr = mem + (size_t)r * 200;
        __builtin_prefetch(mr,       0, 1);
        __builtin_prefetch(mr + 32,  0, 1);
        __builtin_prefetch(mr + 64,  0, 1);
        __builtin_prefetch(mr + 96,  0, 1);
        __builtin_prefetch(mr + 128, 0, 1);
        __builtin_prefetch(mr + 160, 0, 1);
        __builtin_prefetch(mr + 192, 0, 1);
    }

    const int wave = t >> 5;
    const int lane = t & 31;
    const int lo   = lane & 15;   // M (A/D) / N (B/D) within half-wave
    const int hi   = lane >> 4;   // 0: base half, 1: K+2 / M+8 half

    __syncthreads();

    // ---- Fused GEMM: (16 x 8) @ (8 x 272) per row-group, two K=4 steps ----
    for (int gg = 0; gg < 2; ++gg) {
        const int g        = wave * 2 + gg;
        const int rowLocal = g * 16 + lo;
        const int r        = min(rowBase + rowLocal, Btot - 1);
        const float* xr    = x + (size_t)r * 6;

        // A fragment, K step 0: lane-dependent direct load (rows {0,1} / {2,3})
        float2 a01 = *(const float2*)(xr + 2 * hi);
        // A fragment, K step 1: rows {4,5} for lo half; homogeneous {1,0} for hi half
        float2 a45 = *(const float2*)(xr + 4);
        v2f A0 = { a01.x, a01.y };
        v2f A1 = { hi ? 1.0f : a45.x, hi ? 0.0f : a45.y };

        #pragma unroll
        for (int j = 0; j < NTILES; ++j) {
            const float4 q = ldsB[j * 32 + lane];   // one ds_load_b128: both B fragments
            v2f B0 = { q.x, q.y };
            v2f B1 = { q.z, q.w };
            v8f acc = {0.f, 0.f, 0.f, 0.f, 0.f, 0.f, 0.f, 0.f};
            acc = __builtin_amdgcn_wmma_f32_16x16x4_f32(false, A0, false, B0, (short)0, acc, false, false);
            acc = __builtin_amdgcn_wmma_f32_16x16x4_f32(false, A1, false, B1, (short)0, acc, false, false);
            // D layout: VGPR v -> (M=v, N=lo) lanes0-15, (M=v+8, N=lo) lanes16-31
            const int col = j * 16 + lo;
            #pragma unroll
            for (int v = 0; v < 8; ++v)
                vals[(g * 16 + v + 8 * hi) * VSTRIDE + col] = acc[v];
        }
    }
    __syncthreads();

    // --------------------- per-row epilogue (1 thread = 1 row) -------------
    const int row = rowBase + t;
    if (row >= Btot) return;
    const float* vr   = &vals[(size_t)t * VSTRIDE];
    const float* mrow = mem + (size_t)row * 200;

    float c[6];
    {
        float4 c03 = lds4(vr);
        c[0] = c03.x; c[1] = c03.y; c[2] = c03.z; c[3] = c03.w;
        c[4] = vr[4]; c[5] = vr[5];
    }
    const float beta_r = softplusf(vr[COL_BETA_R]);
    const float g_r    = sigm(vr[COL_G_R]);
    const float beta_w = softplusf(vr[COL_BETA_W]);
    const float g_w    = sigm(vr[COL_G_W]);

    float kr[20], kw[20];
    #pragma unroll
    for (int q = 0; q < 5; ++q) {
        float4 fr = lds4(vr + COL_KR + 4 * q);
        float4 fw = lds4(vr + COL_KW + 4 * q);
        kr[4*q] = fr.x; kr[4*q+1] = fr.y; kr[4*q+2] = fr.z; kr[4*q+3] = fr.w;
        kw[4*q] = fw.x; kw[4*q+1] = fw.y; kw[4*q+2] = fw.z; kw[4*q+3] = fw.w;
    }

    float kn_r = 0.0f, kn_w = 0.0f;
    #pragma unroll
    for (int m = 0; m < 20; ++m) { kn_r += kr[m] * kr[m]; kn_w += kw[m] * kw[m]; }
    kn_r = sqrtf(kn_r); kn_w = sqrtf(kn_w);

    // Pass 1 over memory rows: norms + dots -> beta*cos scores
    float sc_r[10], sc_w[10];
    #pragma unroll
    for (int n = 0; n < 10; ++n) {
        float v[20];
        const float4* m4 = (const float4*)(mrow + n * 20);
        #pragma unroll
        for (int q = 0; q < 5; ++q) {
            float4 f = m4[q];
            v[4*q] = f.x; v[4*q+1] = f.y; v[4*q+2] = f.z; v[4*q+3] = f.w;
        }
        float ss = 0.0f, dr = 0.0f, dw = 0.0f;
        #pragma unroll
        for (int m = 0; m < 20; ++m) { ss += v[m]*v[m]; dr += v[m]*kr[m]; dw += v[m]*kw[m]; }
        float mn = sqrtf(ss);
        sc_r[n] = beta_r * (dr / fmaxf(mn * kn_r, 1e-8f));
        sc_w[n] = beta_w * (dw / fmaxf(mn * kn_w, 1e-8f));
    }

    // Softmax + gated interpolation (reference feeds prev_read to BOTH heads)
    float mxr = sc_r[0], mxw = sc_w[0];
    #pragma unroll
    for (int n = 1; n < 10; ++n) { mxr = fmaxf(mxr, sc_r[n]); mxw = fmaxf(mxw, sc_w[n]); }
    float sumr = 0.0f, sumw = 0.0f;
    #pragma unroll
    for (int n = 0; n < 10; ++n) {
        sc_r[n] = __expf(sc_r[n] - mxr); sumr += sc_r[n];
        sc_w[n] = __expf(sc_w[n] - mxw); sumw += sc_w[n];
    }
    const float isr = 1.0f / sumr, isw = 1.0f / sumw;

    const float* pr = prev_read + (size_t)row * 10;
    float st_r[10], st_w[10];
    #pragma unroll
    for (int n = 0; n < 10; ++n) {
        float p = pr[n];
        st_r[n] = g_r * (sc_r[n] * isr) + (1.0f - g_r) * p;
        st_w[n] = g_w * (sc_w[n] * isw) + (1.0f - g_w) * p;
    }

    // Output layout: out(B,6) | state_r(B,10) | state_w(B,10) | new_mem(B,10,20)
    float* o_sr = out + (size_t)Btot * 6  + (size_t)row * 10;
    float* o_sw = out + (size_t)Btot * 16 + (size_t)row * 10;
    #pragma unroll
    for (int n = 0; n < 10; ++n) { o_sr[n] = st_r[n]; o_sw[n] = st_w[n]; }

    // Erase/add vectors: we[m] = sum_n state_w[n] * sigmoid(e_pre[n][m])
    float av[20], we[20];
    #pragma unroll
    for (int q = 0; q < 5; ++q) {
        float4 fa = lds4(vr + COL_A + 4 * q);
        av[4*q] = fa.x; av[4*q+1] = fa.y; av[4*q+2] = fa.z; av[4*q+3] = fa.w;
        we[4*q] = 0.f;  we[4*q+1] = 0.f;  we[4*q+2] = 0.f;  we[4*q+3] = 0.f;
    }
    #pragma unroll
    for (int n = 0; n < 10; ++n) {
        const float sw = st_w[n];
        const float* er = vr + COL_E + n * 20;
        #pragma unroll
        for (int q = 0; q < 5; ++q) {
            float4 f = lds4(er + 4 * q);
            we[4*q]   += sw * sigm(f.x);
            we[4*q+1] += sw * sigm(f.y);
            we[4*q+2] += sw * sigm(f.z);
            we[4*q+3] += sw * sigm(f.w);
        }
    }

    // Pass 2 over memory rows (L2-resident): read vector + memory update
    float rd[20];
    #pragma unroll
    for (int m = 0; m < 20; ++m) rd[m] = 0.0f;
    float* o_nm = out + (size_t)Btot * 26 + (size_t)row * 200;
    #pragma unroll
    for (int n = 0; n < 10; ++n) {
        float v[20];
        const float4* m4 = (const float4*)(mrow + n * 20);
        #pragma unroll
        for (int q = 0; q < 5; ++q) {
            float4 f = m4[q];
            v[4*q] = f.x; v[4*q+1] = f.y; v[4*q+2] = f.z; v[4*q+3] = f.w;
        }
        const float sr = st_r[n], sw = st_w[n];
        float o[20];
        #pragma unroll
        for (int m = 0; m < 20; ++m) {
            rd[m] += sr * v[m];
            o[m] = v[m] * (1.0f - we[m]) + sw * av[m];
        }
        float4* d4 = (float4*)(o_nm + n * 20);
        #pragma unroll
        for (int q = 0; q < 5; ++q) {
            float4 f; f.x = o[4*q]; f.y = o[4*q+1]; f.z = o[4*q+2]; f.w = o[4*q+3];
            d4[q] = f;
        }
    }

    // Final fc: sigmoid([c, read_r] @ Wfc + bfc)
    float inv[26];
    #pragma unroll
    for (int i = 0; i < 6; ++i)  inv[i] = c[i];
    #pragma unroll
    for (int m = 0; m < 20; ++m) inv[6 + m] = rd[m];
    float* o_out = out + (size_t)row * 6;
    #pragma unroll
    for (int o = 0; o < 6; ++o) {
        float acc = bfc[o];
        #pragma unroll
        for (int i = 0; i < 26; ++i) acc += inv[i] * Wfc[i * 6 + o];
        o_out[o] = sigm(acc);
    }
}

extern "C" void kernel_launch(void* const* d_in, const int* in_sizes, int n_in,
                              void* d_out, int out_size, void* d_ws, size_t ws_size,
                              hipStream_t stream) {
    const float* x         = (const float*)d_in[0];
    const float* prev_read = (const float*)d_in[1];
    // d_in[2] = prev_write: dead input (reference passes prev_read to both heads)
    const float* memory    = (const float*)d_in[3];
    const float* Wc  = (const float*)d_in[4],  *bc  = (const float*)d_in[5];
    const float* rWk = (const float*)d_in[6],  *rbk = (const float*)d_in[7];
    const float* rWb = (const float*)d_in[8],  *rbb = (const float*)d_in[9];
    const float* rWg = (const float*)d_in[10], *rbg = (const float*)d_in[11];
    const float* wWk = (const float*)d_in[12], *wbk = (const float*)d_in[13];
    const float* wWb = (const float*)d_in[14], *wbb = (const float*)d_in[15];
    const float* wWg = (const float*)d_in[16], *wbg = (const float*)d_in[17];
    const float* We  = (const float*)d_in[18], *be  = (const float*)d_in[19];
    const float* Wa  = (const float*)d_in[20], *ba  = (const float*)d_in[21];
    const float* Wfc = (const float*)d_in[22], *bfc = (const float*)d_in[23];

    float4* Wpacked = (float4*)d_ws;   // NTILES*32 pre-swizzled B-operand slots

    ntm_fold_weights<<<1, 288, 0, stream>>>(Wc, bc, rWk, rbk, rWb, rbb, rWg, rbg,
                                            wWk, wbk, wWb, wbb, wWg, wbg,
                                            We, be, Wa, ba, Wpacked);

    const int Btot = in_sizes[0] / 6;
    const int grid = (Btot + ROWS_PER_BLOCK - 1) / ROWS_PER_BLOCK;
    const size_t smem_bytes = (size_t)(NTILES * 32 * 4 + ROWS_PER_BLOCK * VSTRIDE) * sizeof(float);

    ntm_main<<<grid, ROWS_PER_BLOCK, smem_bytes, stream>>>(
        x, prev_read, memory, Wpacked, Wfc, bfc, (float*)d_out, Btot);
}